// conv3d_42700564857380
// MI455X (gfx1250) — compile-verified
//
#include <hip/hip_runtime.h>

typedef float v2f __attribute__((ext_vector_type(2)));
typedef float v8f __attribute__((ext_vector_type(8)));

#define KVOL   27
#define MPAIR  50000
#define CIN    64
#define COUT   64

__global__ void zero_out_kernel(float* __restrict__ out, int n) {
    int i = blockIdx.x * blockDim.x + threadIdx.x;
    if (i < n) out[i] = 0.0f;
}

// One block: offset k = blockIdx.x, 8 waves, weights for offset k staged in LDS.
// Each wave processes 16-pair tiles: prefetch next tile's feature rows
// (global_prefetch_b8), gather A (fp32), 64x v_wmma_f32_16x16x4_f32,
// scatter-add via global_atomic_add_f32.
__launch_bounds__(256, 2)
__global__ void sparse_conv_wmma(const float* __restrict__ feats,
                                 const int*   __restrict__ maps,
                                 const float* __restrict__ kernelw,
                                 float*       __restrict__ out) {
    __shared__ float sW[CIN * COUT]; // 16 KB: kernel[k] (Cin x Cout, row-major)

    const int k = blockIdx.x;

    // Cooperative stage of this offset's 64x64 weight into LDS (float4 copies).
    const float4* Wg4 = (const float4*)(kernelw + (size_t)k * CIN * COUT);
    float4* sW4 = (float4*)sW;
    for (int i = threadIdx.x; i < (CIN * COUT) / 4; i += blockDim.x) {
        sW4[i] = Wg4[i];
    }
    __syncthreads();

    const int lane = threadIdx.x & 31;
    const int wave = threadIdx.x >> 5;
    const int half = lane >> 4;   // 0: lanes 0-15, 1: lanes 16-31
    const int l16  = lane & 15;

    const int* mk = maps + (size_t)k * MPAIR * 2;   // [M][2] pairs for offset k

    const int ntiles = MPAIR / 16;                  // 3125, exact
    const int stride = gridDim.y * 8;
    for (int tile = blockIdx.y * 8 + wave; tile < ntiles; tile += stride) {
        const int mbase = tile * 16;

        // ---- Prefetch next tile's gather rows while this tile computes. ----
        {
            int nt = tile + stride;
            if (nt >= ntiles) nt = tile;            // clamp: keep map read in-bounds
            const int nrow = mk[(size_t)(nt * 16 + l16) * 2 + 0];
            __builtin_prefetch(feats + (size_t)nrow * CIN + 2 * half, 0, 0);
        }

        // ---- Gather A: lane covers matrix row m = l16 (pair mbase+l16). ----
        // A-frag layout (16x4 f32): lane holds K = 4*kk + 2*half + {0,1}.
        const int in_row = mk[(size_t)(mbase + l16) * 2 + 0];
        const float* arow = feats + (size_t)in_row * CIN + 2 * half;
        v2f a[16];
#pragma unroll
        for (int kk = 0; kk < 16; ++kk) {
            a[kk].x = arow[4 * kk + 0];
            a[kk].y = arow[4 * kk + 1];
        }

        v8f acc0 = {}, acc1 = {}, acc2 = {}, acc3 = {};

#pragma unroll
        for (int kk = 0; kk < 16; ++kk) {
            // B-frag (4x16 f32): lane holds W[4*kk + 2*half + {0,1}][16*nt + l16]
            const float* wb = sW + (size_t)(4 * kk + 2 * half) * COUT + l16;
            v2f b0, b1, b2, b3;
            b0.x = wb[0];        b0.y = wb[COUT];
            b1.x = wb[16];       b1.y = wb[16 + COUT];
            b2.x = wb[32];       b2.y = wb[32 + COUT];
            b3.x = wb[48];       b3.y = wb[48 + COUT];
            acc0 = __builtin_amdgcn_wmma_f32_16x16x4_f32(false, a[kk], false, b0,
                                                         (short)0, acc0, false, false);
            acc1 = __builtin_amdgcn_wmma_f32_16x16x4_f32(false, a[kk], false, b1,
                                                         (short)0, acc1, false, false);
            acc2 = __builtin_amdgcn_wmma_f32_16x16x4_f32(false, a[kk], false, b2,
                                                         (short)0, acc2, false, false);
            acc3 = __builtin_amdgcn_wmma_f32_16x16x4_f32(false, a[kk], false, b3,
                                                         (short)0, acc3, false, false);
        }

        // ---- Scatter-add: C/D layout -> lane owns rows r + 8*half, col l16. ----
#pragma unroll
        for (int r = 0; r < 8; ++r) {
            const int row  = r + 8 * half;
            const int orow = mk[(size_t)(mbase + row) * 2 + 1];
            float* obase = out + (size_t)orow * COUT + l16;
            atomicAdd(obase + 0,  acc0[r]);
            atomicAdd(obase + 16, acc1[r]);
            atomicAdd(obase + 32, acc2[r]);
            atomicAdd(obase + 48, acc3[r]);
        }
    }
}

extern "C" void kernel_launch(void* const* d_in, const int* in_sizes, int n_in,
                              void* d_out, int out_size, void* d_ws, size_t ws_size,
                              hipStream_t stream) {
    // setup_inputs order: coords (unused), feats, maps, kernel
    const float* feats   = (const float*)d_in[1];
    const int*   maps    = (const int*)d_in[2];
    const float* kernelw = (const float*)d_in[3];
    float* out = (float*)d_out;

    zero_out_kernel<<<(out_size + 255) / 256, 256, 0, stream>>>(out, out_size);

    dim3 grid(KVOL, 48);  // 27 offsets x 48 tile-groups; 8 waves/block
    sparse_conv_wmma<<<grid, dim3(256), 0, stream>>>(feats, maps, kernelw, out);
}